// GraphConv_309237645951
// MI455X (gfx1250) — compile-verified
//
#include <hip/hip_runtime.h>
#include <stdint.h>

#define D    128
#define NREL 20

typedef __attribute__((address_space(3))) float lds_f32_t;

__global__ __launch_bounds__(256) void k_zero4(float4* __restrict__ p, long n4) {
  long i = (long)blockIdx.x * blockDim.x + threadIdx.x;
  long stride = (long)gridDim.x * blockDim.x;
  float4 z = make_float4(0.f, 0.f, 0.f, 0.f);
  for (; i < n4; i += stride) p[i] = z;
}

__global__ __launch_bounds__(256) void k_count(const int* __restrict__ head,
                                               float* __restrict__ cnt, int E) {
  int e = blockIdx.x * blockDim.x + threadIdx.x;
  if (e < E) atomicAdd(&cnt[head[e]], 1.0f);
}

__global__ __launch_bounds__(256) void k_init(const float4* __restrict__ ego,
                                              float4* __restrict__ x,
                                              float4* __restrict__ res, long n4) {
  long i = (long)blockIdx.x * blockDim.x + threadIdx.x;
  if (i < n4) {
    float4 v = ego[i];
    x[i] = v;
    res[i] = v;
  }
}

// One wave32 per edge; lane l owns features [4l, 4l+4). Relation table staged
// into LDS with CDNA5 async global->LDS loads (ASYNCcnt).
__global__ __launch_bounds__(256) void k_edge(const float* __restrict__ x,
                                              const float* __restrict__ rel,
                                              const int* __restrict__ head,
                                              const int* __restrict__ tail,
                                              const int* __restrict__ etype,
                                              float* __restrict__ agg, int E) {
  __shared__ float s_rel[NREL * D];  // 10 KB of 320 KB WGP LDS

  const int tid = threadIdx.x;
  // Async-stage relation_embed (20x128 f32) into LDS: each lane issues a
  // global_load_async_to_lds_b128 with its own LDS byte address (vdst) and
  // 32-bit global byte offset (vaddr) against the uniform SGPR base (saddr).
  for (int j = tid; j < (NREL * D) / 4; j += blockDim.x) {
    unsigned lds_off = (unsigned)(uintptr_t)(lds_f32_t*)(&s_rel[4 * j]);
    unsigned goff = (unsigned)(16u * (unsigned)j);
    asm volatile("global_load_async_to_lds_b128 %0, %1, %2"
                 :
                 : "v"(lds_off), "v"(goff), "s"(rel)
                 : "memory");
  }
  asm volatile("s_wait_asynccnt 0" ::: "memory");
  __syncthreads();

  const int lane = tid & 31;
  const int wpb = blockDim.x >> 5;
  const int stride = gridDim.x * wpb;
  const int c = lane * 4;

  for (int e = blockIdx.x * wpb + (tid >> 5); e < E; e += stride) {
    // Prefetch next iteration's tail row into cache (global_prefetch_b8).
    const int en = e + stride;
    if (en < E) {
      const int tn = tail[en];
      __builtin_prefetch(&x[(size_t)tn * D + c], 0, 3);
    }

    const int h = head[e];
    const int t = tail[e];
    const int r = etype[e];

    const float4 xv = *(const float4*)(x + (size_t)t * D + c);
    const float4 rv = *(const float4*)(s_rel + r * D + c);

    float* a = agg + (size_t)h * D + c;
    atomicAdd(a + 0, xv.x * rv.x);
    atomicAdd(a + 1, xv.y * rv.y);
    atomicAdd(a + 2, xv.z * rv.z);
    atomicAdd(a + 3, xv.w * rv.w);
  }
}

// One wave32 per node row: scatter-mean, L2 normalize (wave butterfly
// reduction), write new x, accumulate residual into res (d_out).
__global__ __launch_bounds__(256) void k_norm(const float* __restrict__ agg,
                                              const float* __restrict__ cnt,
                                              float* __restrict__ x,
                                              float* __restrict__ res, int N) {
  const int tid = threadIdx.x;
  const int lane = tid & 31;
  const int n = blockIdx.x * (blockDim.x >> 5) + (tid >> 5);
  if (n >= N) return;

  const float rcp = 1.0f / fmaxf(cnt[n], 1.0f);
  const size_t base = (size_t)n * D + (size_t)(lane * 4);

  float4 v = *(const float4*)(agg + base);
  v.x *= rcp; v.y *= rcp; v.z *= rcp; v.w *= rcp;

  float ss = v.x * v.x + v.y * v.y + v.z * v.z + v.w * v.w;
#pragma unroll
  for (int m = 16; m >= 1; m >>= 1) ss += __shfl_xor(ss, m, 32);

  const float scale = 1.0f / fmaxf(sqrtf(ss), 1e-12f);
  v.x *= scale; v.y *= scale; v.z *= scale; v.w *= scale;

  *(float4*)(x + base) = v;

  float4 o = *(const float4*)(res + base);
  o.x += v.x; o.y += v.y; o.z += v.z; o.w += v.w;
  *(float4*)(res + base) = o;
}

extern "C" void kernel_launch(void* const* d_in, const int* in_sizes, int n_in,
                              void* d_out, int out_size, void* d_ws, size_t ws_size,
                              hipStream_t stream) {
  (void)n_in; (void)out_size; (void)ws_size;

  const float* ego  = (const float*)d_in[0];
  const int*   eidx = (const int*)d_in[1];
  const int*   etyp = (const int*)d_in[2];
  const float* rel  = (const float*)d_in[3];
  // d_in[4] = dropout (0 in eval mode) -> ignored

  const int E = in_sizes[2];        // edge count
  const int N = in_sizes[0] / D;    // node count
  const int* head = eidx;           // edge_index[0]
  const int* tail = eidx + E;       // edge_index[1]

  // Workspace: cnt[N] | x[N*D] | agg[N*D]
  float* cnt = (float*)d_ws;
  float* x   = cnt + N;
  float* agg = x + (size_t)N * D;
  float* res = (float*)d_out;

  const long nodes4 = (long)N * D / 4;

  // In-degree counts (constant across hops)
  k_zero4<<<(N / 4 + 255) / 256, 256, 0, stream>>>((float4*)cnt, (long)(N / 4));
  k_count<<<(E + 255) / 256, 256, 0, stream>>>(head, cnt, E);

  // res = ego, x = ego
  k_init<<<(int)((nodes4 + 255) / 256), 256, 0, stream>>>(
      (const float4*)ego, (float4*)x, (float4*)res, nodes4);

  for (int hop = 0; hop < 3; ++hop) {
    k_zero4<<<4096, 256, 0, stream>>>((float4*)agg, nodes4);
    k_edge<<<4096, 256, 0, stream>>>(x, rel, head, tail, etyp, agg, E);
    k_norm<<<(N + 7) / 8, 256, 0, stream>>>(agg, cnt, x, res, N);
  }
}